// OuterProduct_4629974745109
// MI455X (gfx1250) — compile-verified
//
#include <hip/hip_runtime.h>

typedef __attribute__((ext_vector_type(2))) float v2f;
typedef __attribute__((ext_vector_type(8))) float v8f;

#define B_  8
#define C_  64
#define L_  384
#define TILES_1D (L_ / 16)                 // 24 sixteen-wide tiles per dim
#define JT  4                              // j sub-tiles per wave (16x64 tile)
#define TILES_J (TILES_1D / JT)            // 6 wave-tiles along j
#define TILES_PW (TILES_1D * TILES_J)      // 144 wave-tiles per (b,c)
#define BC_ (B_ * C_)                      // 512 (b,c) pairs
#define WAVES_PER_BLOCK 8
#define TOTAL_WAVES (BC_ * TILES_PW)       // 73728
#define NUM_BLOCKS (TOTAL_WAVES / WAVES_PER_BLOCK) // 9216

// One wave32 computes a 16x64 (i,j) tile for one (b,c):
//   product plane: 4x V_WMMA_F32_16X16X4_F32 (rank-1 matmul, K padded to 4),
//                  A operand (s1 column) loaded once and reused
//   abs-diff plane: VALU
// All 128 outputs-per-row-pair stored via NT b32 stores using only two base
// pointers + immediate offsets (r*1536 + q*64 bytes, fits IOFFSET).
__global__ __launch_bounds__(256) void
outer_pair_kernel(const float* __restrict__ s1,
                  const float* __restrict__ s2,
                  float* __restrict__ out)
{
    const int lane = threadIdx.x & 31;
    const int wave = threadIdx.x >> 5;

    const unsigned gw = blockIdx.x * (unsigned)WAVES_PER_BLOCK + wave;
    const unsigned bc = gw / TILES_PW;          // which (b,c) pair
    const unsigned t  = gw % TILES_PW;
    const unsigned ti = t / TILES_J;
    const unsigned tj = t % TILES_J;
    const unsigned b  = bc / C_;
    const unsigned c  = bc % C_;
    const unsigned i0 = ti * 16;
    const unsigned j0 = tj * 64;

    const float* __restrict__ s1p = s1 + (size_t)bc * L_;
    const float* __restrict__ s2p = s2 + (size_t)bc * L_;

    const int  nlo     = lane & 15;
    const bool lowHalf = lane < 16;
    const int  rbase   = lowHalf ? 0 : 8;       // D rows 0-7 / 8-15 per half-wave

    // ---- WMMA A operand (16x4, only K=0 column non-zero): loaded once ----
    v2f A;
    A[0] = lowHalf ? s1p[i0 + nlo] : 0.0f;
    A[1] = 0.0f;

    // ---- s1 broadcast values for the abs tile: loaded once ----
    float s1row[8];
#pragma unroll
    for (int r = 0; r < 8; ++r)
        s1row[r] = s1p[i0 + rbase + r];

    // ---- output base pointers (abs plane c, product plane C+c) ----
    const size_t planeAbs  = (size_t)b * (2 * C_) + c;
    const size_t planeProd = planeAbs + C_;
    float* pa = out + (planeAbs  * L_ + (i0 + rbase)) * (size_t)L_ + j0 + nlo;
    float* pp = out + (planeProd * L_ + (i0 + rbase)) * (size_t)L_ + j0 + nlo;

    v8f Cz = {};

#pragma unroll
    for (int q = 0; q < JT; ++q) {
        // s2 value for this 16-column strip (same for both half-waves)
        const float bq = s2p[j0 + q * 16 + nlo];

        // B operand (4x16, only K=0 row non-zero)
        v2f Bm;
        Bm[0] = lowHalf ? bq : 0.0f;
        Bm[1] = 0.0f;

        // (neg_a, A, neg_b, B, c_mod, C, reuse_a, reuse_b)
        v8f D = __builtin_amdgcn_wmma_f32_16x16x4_f32(
            false, A, false, Bm, (short)0, Cz, false, false);

#pragma unroll
        for (int r = 0; r < 8; ++r) {
            __builtin_nontemporal_store(__builtin_fabsf(s1row[r] - bq),
                                        pa + (size_t)r * L_ + q * 16);
            __builtin_nontemporal_store(D[r],
                                        pp + (size_t)r * L_ + q * 16);
        }
    }
}

extern "C" void kernel_launch(void* const* d_in, const int* in_sizes, int n_in,
                              void* d_out, int out_size, void* d_ws, size_t ws_size,
                              hipStream_t stream)
{
    const float* seq1 = (const float*)d_in[0];
    const float* seq2 = (const float*)d_in[1];
    float* out = (float*)d_out;

    dim3 grid(NUM_BLOCKS);
    dim3 block(WAVES_PER_BLOCK * 32);
    outer_pair_kernel<<<grid, block, 0, stream>>>(seq1, seq2, out);
}